// ChainVigilGNN_82970178224651
// MI455X (gfx1250) — compile-verified
//
#include <hip/hip_runtime.h>
#include <hip/hip_bf16.h>
#include <math.h>

typedef float v2f __attribute__((ext_vector_type(2)));
typedef float v8f __attribute__((ext_vector_type(8)));

#define HDIM   128
#define HEADS  4
#define CH     32
#define LDT    132           // padded LDS pitch: lane stride 132*4B -> 4 banks apart, conflict-free
#define BN_EPS 1e-5f
#define NEG_SLOPE 0.2f

// ---------------------------------------------------------------------------
// order-preserving float <-> uint mapping for atomicMax on floats
// ---------------------------------------------------------------------------
__device__ __forceinline__ unsigned f2ord(float f) {
    unsigned u = __float_as_uint(f);
    return (u & 0x80000000u) ? ~u : (u | 0x80000000u);
}
__device__ __forceinline__ float ord2f(unsigned u) {
    return __uint_as_float((u & 0x80000000u) ? (u & 0x7FFFFFFFu) : ~u);
}

// ---------------------------------------------------------------------------
// one 16x16 WMMA tile of  D += A(16x128, LDS, pitch LDT) * B(128x128, global)
// V_WMMA_F32_16X16X4_F32: A frag lane l -> row (l&15), K = 2*(l>>4)+{0,1}
//                         B frag lane l -> col (l&15), same K split
// ---------------------------------------------------------------------------
__device__ __forceinline__ v8f wmma_tile_k128(const float* __restrict__ sA,
                                              const float* __restrict__ B,
                                              int lane, int col0, v8f acc) {
    const int m  = lane & 15;
    const int kh = (lane >> 4) << 1;       // 0 or 2
    const int n  = col0 + m;
    const float* ar = sA + m * LDT;
#pragma unroll
    for (int k0 = 0; k0 < HDIM; k0 += 4) {
        v2f a, b;
        a.x = ar[k0 + kh];
        a.y = ar[k0 + kh + 1];
        b.x = B[(size_t)(k0 + kh)     * HDIM + n];
        b.y = B[(size_t)(k0 + kh + 1) * HDIM + n];
        acc = __builtin_amdgcn_wmma_f32_16x16x4_f32(false, a, false, b,
                                                    (short)0, acc, false, false);
    }
    return acc;
}

// ---------------------------------------------------------------------------
// Edge scatter:  agg[dst] += x[src]  (128 floats),  deg[dst] += 1
// wave (32 lanes) per edge, 4 floats per lane
// ---------------------------------------------------------------------------
__global__ __launch_bounds__(256) void scatter_agg_k(const float* __restrict__ x,
                                                     const long long* __restrict__ ei,
                                                     float* __restrict__ agg,
                                                     float* __restrict__ deg,
                                                     int E) {
    int tid  = blockIdx.x * 256 + threadIdx.x;
    int e    = tid >> 5;
    int lane = tid & 31;
    if (e >= E) return;
    int s = (int)ei[e];
    int d = (int)ei[(size_t)E + e];
    float4 v = *(const float4*)(x + (size_t)s * HDIM + lane * 4);
    float* o = agg + (size_t)d * HDIM + lane * 4;
    atomicAdd(o + 0, v.x);
    atomicAdd(o + 1, v.y);
    atomicAdd(o + 2, v.z);
    atomicAdd(o + 3, v.w);
    if (lane == 0) atomicAdd(&deg[d], 1.0f);
}

// ---------------------------------------------------------------------------
// SAGE:  h1 = relu(bn1( (agg/deg) @ wl + bl + x @ wr ))       [WMMA f32]
// block = 256 thr = 8 waves, computes 16 rows x 128 cols
// ---------------------------------------------------------------------------
__global__ __launch_bounds__(256) void sage_k(const float* __restrict__ x,
                                              const float* __restrict__ agg,
                                              const float* __restrict__ deg,
                                              const float* __restrict__ wl,
                                              const float* __restrict__ bl,
                                              const float* __restrict__ wr,
                                              const float* __restrict__ g,
                                              const float* __restrict__ be,
                                              const float* __restrict__ mu,
                                              const float* __restrict__ var,
                                              float* __restrict__ h1) {
    __shared__ float sA[16 * LDT];
    __shared__ float sX[16 * LDT];
    const int t    = threadIdx.x;
    const int row0 = blockIdx.x * 16;
    // stage 16x128 tiles (coalesced float4), fold mean-divide into the stage
#pragma unroll
    for (int i = 0; i < 2; ++i) {
        int f  = i * 256 + t;            // 512 float4 slots
        int r  = f >> 5;                 // 0..15
        int c  = (f & 31) * 4;           // 0..124
        float dinv = 1.0f / fmaxf(deg[row0 + r], 1.0f);
        float4 va = *(const float4*)(agg + (size_t)(row0 + r) * HDIM + c);
        float4 vx = *(const float4*)(x   + (size_t)(row0 + r) * HDIM + c);
        float* pa = sA + r * LDT + c;
        float* px = sX + r * LDT + c;
        pa[0] = va.x * dinv; pa[1] = va.y * dinv; pa[2] = va.z * dinv; pa[3] = va.w * dinv;
        px[0] = vx.x;        px[1] = vx.y;        px[2] = vx.z;        px[3] = vx.w;
    }
    __syncthreads();
    const int wave = t >> 5, lane = t & 31;
    const int col0 = wave * 16;
    v8f acc = {};
    acc = wmma_tile_k128(sA, wl, lane, col0, acc);   // agg @ wl
    acc = wmma_tile_k128(sX, wr, lane, col0, acc);   // + x @ wr
    // epilogue: + bias, BN1, ReLU
    const int n     = col0 + (lane & 15);
    const int mbase = (lane >> 4) * 8;
    const float scale = g[n] * rsqrtf(var[n] + BN_EPS);
    const float mm = mu[n], bb = be[n], bias = bl[n];
#pragma unroll
    for (int r = 0; r < 8; ++r) {
        float v = acc[r] + bias;
        v = (v - mm) * scale + bb;
        h1[(size_t)(row0 + mbase + r) * HDIM + n] = fmaxf(v, 0.0f);
    }
}

// ---------------------------------------------------------------------------
// GAT transform: xh = h1 @ gat_w                              [WMMA f32]
// ---------------------------------------------------------------------------
__global__ __launch_bounds__(256) void gatw_k(const float* __restrict__ h1,
                                              const float* __restrict__ gw,
                                              float* __restrict__ xh) {
    __shared__ float sA[16 * LDT];
    const int t    = threadIdx.x;
    const int row0 = blockIdx.x * 16;
#pragma unroll
    for (int i = 0; i < 2; ++i) {
        int f = i * 256 + t;
        int r = f >> 5;
        int c = (f & 31) * 4;
        float4 v = *(const float4*)(h1 + (size_t)(row0 + r) * HDIM + c);
        float* p = sA + r * LDT + c;
        p[0] = v.x; p[1] = v.y; p[2] = v.z; p[3] = v.w;
    }
    __syncthreads();
    const int wave = t >> 5, lane = t & 31;
    const int col0 = wave * 16;
    v8f acc = {};
    acc = wmma_tile_k128(sA, gw, lane, col0, acc);
    const int n     = col0 + (lane & 15);
    const int mbase = (lane >> 4) * 8;
#pragma unroll
    for (int r = 0; r < 8; ++r)
        xh[(size_t)(row0 + mbase + r) * HDIM + n] = acc[r];
}

// ---------------------------------------------------------------------------
// per-(node,head) attention coefficients  a_src / a_dst
// ---------------------------------------------------------------------------
__global__ __launch_bounds__(256) void attn_coef_k(const float* __restrict__ xh,
                                                   const float* __restrict__ att_s,
                                                   const float* __restrict__ att_d,
                                                   float* __restrict__ a_src,
                                                   float* __restrict__ a_dst,
                                                   int N) {
    int idx = blockIdx.x * 256 + threadIdx.x;        // N*HEADS threads
    if (idx >= N * HEADS) return;
    int hd = idx & 3, node = idx >> 2;
    const float4* xr = (const float4*)(xh    + (size_t)node * HDIM + hd * CH);
    const float4* as = (const float4*)(att_s + hd * CH);
    const float4* ad = (const float4*)(att_d + hd * CH);
    float s = 0.f, d = 0.f;
#pragma unroll
    for (int i = 0; i < CH / 4; ++i) {
        float4 v = xr[i], a1 = as[i], a2 = ad[i];
        s += v.x * a1.x + v.y * a1.y + v.z * a1.z + v.w * a1.w;
        d += v.x * a2.x + v.y * a2.y + v.z * a2.z + v.w * a2.w;
    }
    a_src[idx] = s;
    a_dst[idx] = d;
}

// ---------------------------------------------------------------------------
// segment max over edges (per dst,head) via ordered-uint atomicMax
// ---------------------------------------------------------------------------
__global__ __launch_bounds__(256) void edge_max_k(const long long* __restrict__ ei,
                                                  const float* __restrict__ a_src,
                                                  const float* __restrict__ a_dst,
                                                  unsigned* __restrict__ emax,
                                                  int E) {
    int idx = blockIdx.x * 256 + threadIdx.x;        // E*HEADS threads
    if (idx >= E * HEADS) return;
    int e = idx >> 2, hd = idx & 3;
    int s = (int)ei[e];
    int d = (int)ei[(size_t)E + e];
    float v = a_src[s * HEADS + hd] + a_dst[d * HEADS + hd];
    v = (v > 0.f) ? v : NEG_SLOPE * v;               // leaky_relu
    atomicMax(&emax[d * HEADS + hd], f2ord(v));
}

// ---------------------------------------------------------------------------
// e_exp = exp(e - max);  esum[dst,head] += e_exp
// ---------------------------------------------------------------------------
__global__ __launch_bounds__(256) void edge_expsum_k(const long long* __restrict__ ei,
                                                     const float* __restrict__ a_src,
                                                     const float* __restrict__ a_dst,
                                                     const unsigned* __restrict__ emax,
                                                     float* __restrict__ eexp,
                                                     float* __restrict__ esum,
                                                     int E) {
    int idx = blockIdx.x * 256 + threadIdx.x;
    if (idx >= E * HEADS) return;
    int e = idx >> 2, hd = idx & 3;
    int s = (int)ei[e];
    int d = (int)ei[(size_t)E + e];
    float v = a_src[s * HEADS + hd] + a_dst[d * HEADS + hd];
    v = (v > 0.f) ? v : NEG_SLOPE * v;
    float ex = __expf(v - ord2f(emax[d * HEADS + hd]));
    eexp[idx] = ex;
    atomicAdd(&esum[d * HEADS + hd], ex);
}

// ---------------------------------------------------------------------------
// out[dst] += alpha * xh[src]   (wave per edge, 4 floats/lane, head = lane>>3)
// ---------------------------------------------------------------------------
__global__ __launch_bounds__(256) void edge_scatter_k(const long long* __restrict__ ei,
                                                      const float* __restrict__ xh,
                                                      const float* __restrict__ eexp,
                                                      const float* __restrict__ esum,
                                                      float* __restrict__ outg,
                                                      int E) {
    int tid  = blockIdx.x * 256 + threadIdx.x;
    int e    = tid >> 5;
    int lane = tid & 31;
    if (e >= E) return;
    int s = (int)ei[e];
    int d = (int)ei[(size_t)E + e];
    int head = lane >> 3;                            // 4 cols/lane stay in one head
    float alpha = eexp[(size_t)e * HEADS + head] /
                  (esum[(size_t)d * HEADS + head] + 1e-16f);
    float4 v = *(const float4*)(xh + (size_t)s * HDIM + lane * 4);
    float* o = outg + (size_t)d * HDIM + lane * 4;
    atomicAdd(o + 0, alpha * v.x);
    atomicAdd(o + 1, alpha * v.y);
    atomicAdd(o + 2, alpha * v.z);
    atomicAdd(o + 3, alpha * v.w);
}

// ---------------------------------------------------------------------------
// h2 = relu(bn2(out + gat_bias))  -> embeddings region of d_out
// ---------------------------------------------------------------------------
__global__ __launch_bounds__(256) void bn2_k(const float* __restrict__ outg,
                                             const float* __restrict__ gb,
                                             const float* __restrict__ g,
                                             const float* __restrict__ be,
                                             const float* __restrict__ mu,
                                             const float* __restrict__ var,
                                             float* __restrict__ emb,
                                             int N) {
    int idx = blockIdx.x * 256 + threadIdx.x;        // N*32 threads
    if (idx >= N * 32) return;
    int n = idx >> 5, c = (idx & 31) * 4;
    float4 v  = *(const float4*)(outg + (size_t)n * HDIM + c);
    float4 bv = *(const float4*)(gb  + c);
    float4 gv = *(const float4*)(g   + c);
    float4 ev = *(const float4*)(be  + c);
    float4 mv = *(const float4*)(mu  + c);
    float4 vv = *(const float4*)(var + c);
    float4 o;
    o.x = fmaxf((v.x + bv.x - mv.x) * (gv.x * rsqrtf(vv.x + BN_EPS)) + ev.x, 0.f);
    o.y = fmaxf((v.y + bv.y - mv.y) * (gv.y * rsqrtf(vv.y + BN_EPS)) + ev.y, 0.f);
    o.z = fmaxf((v.z + bv.z - mv.z) * (gv.z * rsqrtf(vv.z + BN_EPS)) + ev.z, 0.f);
    o.w = fmaxf((v.w + bv.w - mv.w) * (gv.w * rsqrtf(vv.w + BN_EPS)) + ev.w, 0.f);
    *(float4*)(emb + (size_t)n * HDIM + c) = o;
}

// ---------------------------------------------------------------------------
// MLP 128->64->32->1 + sigmoid; weights in LDS (broadcast reads)
// ---------------------------------------------------------------------------
__global__ __launch_bounds__(256) void mlp_k(const float* __restrict__ emb,
                                             const float* __restrict__ w1,
                                             const float* __restrict__ b1,
                                             const float* __restrict__ w2,
                                             const float* __restrict__ b2,
                                             const float* __restrict__ w3,
                                             const float* __restrict__ b3,
                                             float* __restrict__ probs,
                                             int N) {
    __shared__ float sw1[128 * 64];
    __shared__ float sw2[64 * 32];
    __shared__ float sb1[64];
    __shared__ float sb2[32];
    __shared__ float sw3[32];
    int t = threadIdx.x;
    for (int i = t; i < 128 * 64; i += 256) sw1[i] = w1[i];
    for (int i = t; i < 64 * 32;  i += 256) sw2[i] = w2[i];
    if (t < 64) sb1[t] = b1[t];
    if (t < 32) { sb2[t] = b2[t]; sw3[t] = w3[t]; }
    __syncthreads();
    int node = blockIdx.x * 256 + t;
    if (node >= N) return;

    float z1[64];
#pragma unroll
    for (int j = 0; j < 64; ++j) z1[j] = sb1[j];
    for (int k = 0; k < 128; k += 4) {
        float4 h = *(const float4*)(emb + (size_t)node * HDIM + k);
#pragma unroll
        for (int j = 0; j < 64; ++j)
            z1[j] += h.x * sw1[(k + 0) * 64 + j] + h.y * sw1[(k + 1) * 64 + j]
                   + h.z * sw1[(k + 2) * 64 + j] + h.w * sw1[(k + 3) * 64 + j];
    }
#pragma unroll
    for (int j = 0; j < 64; ++j) z1[j] = fmaxf(z1[j], 0.f);

    float z2[32];
#pragma unroll
    for (int j = 0; j < 32; ++j) z2[j] = sb2[j];
    for (int k = 0; k < 64; ++k) {
        float v = z1[k];
#pragma unroll
        for (int j = 0; j < 32; ++j) z2[j] += v * sw2[k * 32 + j];
    }
    float lg = b3[0];
#pragma unroll
    for (int k = 0; k < 32; ++k) lg += fmaxf(z2[k], 0.f) * sw3[k];
    probs[node] = 1.0f / (1.0f + __expf(-lg));
}

// ---------------------------------------------------------------------------
extern "C" void kernel_launch(void* const* d_in, const int* in_sizes, int n_in,
                              void* d_out, int out_size, void* d_ws, size_t ws_size,
                              hipStream_t stream) {
    const float*     x     = (const float*)d_in[0];
    const long long* ei    = (const long long*)d_in[1];
    const float*     swl   = (const float*)d_in[2];
    const float*     sbl   = (const float*)d_in[3];
    const float*     swr   = (const float*)d_in[4];
    const float*     bn1g  = (const float*)d_in[5];
    const float*     bn1b  = (const float*)d_in[6];
    const float*     bn1m  = (const float*)d_in[7];
    const float*     bn1v  = (const float*)d_in[8];
    const float*     gatw  = (const float*)d_in[9];
    const float*     attS  = (const float*)d_in[10];
    const float*     attD  = (const float*)d_in[11];
    const float*     gatb  = (const float*)d_in[12];
    const float*     bn2g  = (const float*)d_in[13];
    const float*     bn2b  = (const float*)d_in[14];
    const float*     bn2m  = (const float*)d_in[15];
    const float*     bn2v  = (const float*)d_in[16];
    const float*     w1    = (const float*)d_in[17];
    const float*     b1    = (const float*)d_in[18];
    const float*     w2    = (const float*)d_in[19];
    const float*     b2    = (const float*)d_in[20];
    const float*     w3    = (const float*)d_in[21];
    const float*     b3    = (const float*)d_in[22];

    const int N = in_sizes[0] / HDIM;       // 100000
    const int E = in_sizes[1] / 2;          // 1600000

    // workspace layout (floats); bufA: agg -> xh, bufB: h1 -> out_gat
    float* ws    = (float*)d_ws;
    size_t o     = 0;
    float* deg   = ws + o; o += (size_t)N;
    float* bufA  = ws + o; o += (size_t)N * HDIM;
    float* bufB  = ws + o; o += (size_t)N * HDIM;
    float* a_src = ws + o; o += (size_t)N * HEADS;
    float* a_dst = ws + o; o += (size_t)N * HEADS;
    unsigned* emax = (unsigned*)(ws + o); o += (size_t)N * HEADS;
    float* esum  = ws + o; o += (size_t)N * HEADS;
    float* eexp  = ws + o; o += (size_t)E * HEADS;

    float* probs = (float*)d_out;           // [N]
    float* emb   = (float*)d_out + N;       // [N,128]

    // ---- SAGE ----
    hipMemsetAsync(deg,  0, (size_t)N * sizeof(float), stream);
    hipMemsetAsync(bufA, 0, (size_t)N * HDIM * sizeof(float), stream);
    {
        int blocks = (int)(((size_t)E * 32 + 255) / 256);
        scatter_agg_k<<<blocks, 256, 0, stream>>>(x, ei, bufA, deg, E);
    }
    sage_k<<<N / 16, 256, 0, stream>>>(x, bufA, deg, swl, sbl, swr,
                                       bn1g, bn1b, bn1m, bn1v, bufB);

    // ---- GAT ----
    gatw_k<<<N / 16, 256, 0, stream>>>(bufB, gatw, bufA);           // xh = h1 @ W
    attn_coef_k<<<(N * HEADS + 255) / 256, 256, 0, stream>>>(bufA, attS, attD,
                                                             a_src, a_dst, N);
    hipMemsetAsync(emax, 0, (size_t)N * HEADS * sizeof(unsigned), stream);
    hipMemsetAsync(esum, 0, (size_t)N * HEADS * sizeof(float), stream);
    hipMemsetAsync(bufB, 0, (size_t)N * HDIM * sizeof(float), stream); // out_gat
    {
        int blocks = (E * HEADS + 255) / 256;
        edge_max_k<<<blocks, 256, 0, stream>>>(ei, a_src, a_dst, emax, E);
        edge_expsum_k<<<blocks, 256, 0, stream>>>(ei, a_src, a_dst, emax,
                                                  eexp, esum, E);
    }
    {
        int blocks = (int)(((size_t)E * 32 + 255) / 256);
        edge_scatter_k<<<blocks, 256, 0, stream>>>(ei, bufA, eexp, esum, bufB, E);
    }
    bn2_k<<<(N * 32 + 255) / 256, 256, 0, stream>>>(bufB, gatb,
                                                    bn2g, bn2b, bn2m, bn2v,
                                                    emb, N);

    // ---- MLP + sigmoid ----
    mlp_k<<<(N + 255) / 256, 256, 0, stream>>>(emb, w1, b1, w2, b2, w3, b3,
                                               probs, N);
}